// SFM_49100066128224
// MI455X (gfx1250) — compile-verified
//
#include <hip/hip_runtime.h>
#include <hip/hip_bf16.h>
#include <math.h>

#define SFM_DT  0.02f
#define SFM_EPS 1e-8f

typedef float v2f __attribute__((ext_vector_type(2)));
typedef float v8f __attribute__((ext_vector_type(8)));

// ---------------------------------------------------------------------------
// mono MLP for 16 agents per wave via V_WMMA_F32_16X16X4_F32 (K=16 as 4 chunks).
//
// Agent m is owned by lanes m and m+16. A-matrix f32 16x4 layout (ISA 7.12.2):
//   lanes 0-15  : row m=lane,    VGPR0=K(4c+0), VGPR1=K(4c+1)
//   lanes 16-31 : row m=lane-16, VGPR0=K(4c+2), VGPR1=K(4c+3)
// so each lane evaluates h components widx(t)=4*(t>>1)+(t&1)+2*half, t=0..7.
// B (4x16) mirrors A (column-per-lane); we replicate w2 across all columns so
// D[m][n] = b2 + h_m . w2  for every n.  C is preloaded with b2.
// D layout: lanes 0-15 hold rows 0-7 (VGPR r = row r), lanes 16-31 rows 8-15;
// a register select + one shuffle returns agent (lane%16)'s value to its lanes.
// ---------------------------------------------------------------------------
__device__ __forceinline__ float mono16_wmma(float xv,
    const float* w1v, const float* b1v, const float* w2v, float b2, int lane)
{
  float h[8];
#pragma unroll
  for (int t = 0; t < 8; ++t)
    h[t] = __expf(-fmaf(xv, w1v[t], b1v[t]));

  v8f c;
#pragma unroll
  for (int r = 0; r < 8; ++r) c[r] = b2;

#pragma unroll
  for (int ch = 0; ch < 4; ++ch) {
    v2f a, b;
    a[0] = h[2*ch];     a[1] = h[2*ch + 1];
    b[0] = w2v[2*ch];   b[1] = w2v[2*ch + 1];
    // 8 args: (neg_a, A, neg_b, B, c_mod, C, reuse_a, reuse_b)
    c = __builtin_amdgcn_wmma_f32_16x16x4_f32(false, a, false, b,
                                              (short)0, c, false, false);
  }

  // pick VGPR (lane&7) then fix the half (rows 0-7 live in lanes 0-15,
  // rows 8-15 in lanes 16-31; columns are replicated so any lane works).
  int s = lane & 7;
  float t8 = c[0];
  t8 = (s == 1) ? c[1] : t8;
  t8 = (s == 2) ? c[2] : t8;
  t8 = (s == 3) ? c[3] : t8;
  t8 = (s == 4) ? c[4] : t8;
  t8 = (s == 5) ? c[5] : t8;
  t8 = (s == 6) ? c[6] : t8;
  t8 = (s == 7) ? c[7] : t8;
  bool crossed = (((lane & 8) != 0) != ((lane & 16) != 0));
  int src = crossed ? (lane ^ 16) : lane;
  return __shfl(t8, src, 32);
}

// angle clamp, literal reference semantics (NaN compare -> false -> zero)
__device__ __forceinline__ void angle_clamp2(float vx, float vy, float vn,
                                             float ang, float& fx, float& fy)
{
  float num = vx * fx + vy * fy;
  float fn  = sqrtf(fx * fx + fy * fy);
  float den = fmaxf(vn, SFM_EPS) * fmaxf(fn, SFM_EPS);
  bool keep = fabsf(num / den) > ang;
  if (!keep) { fx = 0.0f; fy = 0.0f; }
}

// ---------------------------------------------------------------------------
// Precompute: dur[8] depends only on ego[1,:,8:16]; dtf[j] = mono_dtf(dur[j]).
// One tiny block; result (8 floats) goes to workspace.
// ---------------------------------------------------------------------------
__global__ void sfm_pre(const float* __restrict__ ego,
                        const float* __restrict__ w1, const float* __restrict__ b1,
                        const float* __restrict__ w2, const float* __restrict__ b2,
                        float* __restrict__ ws)
{
  int j = threadIdx.x;
  if (j < 8) {
    const float* base = ego + 68;            // ego[1, :, :], 4*17 floats
    float lastid = base[3 * 17 + 8 + j];
    float dur = 0.0f;
#pragma unroll
    for (int t = 0; t < 4; ++t) {
      bool any = false;
#pragma unroll
      for (int k = 0; k < 8; ++k)
        any = any || (lastid == base[t * 17 + 8 + k]);
      dur += any ? 1.0f : 0.0f;
    }
    float y = b2[0];
#pragma unroll
    for (int i = 0; i < 16; ++i)
      y += w2[i] * __expf(-fmaf(dur, w1[i], b1[i]));
    ws[j] = y;
  }
}

// ---------------------------------------------------------------------------
// Main kernel: 256 threads = 8 waves; 16 agents per wave (agent = lane%16,
// lane pairs cooperate on WMMA rows). 18 WMMA tiles per agent group.
// ---------------------------------------------------------------------------
__global__ __launch_bounds__(256) void sfm_main(
    const float* __restrict__ ego, const float* __restrict__ nei,
    const float* __restrict__ border, const float* __restrict__ adp,
    const float* __restrict__ eang,
    const float* __restrict__ aw1, const float* __restrict__ ab1,
    const float* __restrict__ aw2, const float* __restrict__ ab2,
    const float* __restrict__ rw1, const float* __restrict__ rb1,
    const float* __restrict__ rw2, const float* __restrict__ rb2,
    const float* __restrict__ bw1, const float* __restrict__ bb1,
    const float* __restrict__ bw2, const float* __restrict__ bb2,
    const float* __restrict__ dtf, float* __restrict__ out, int B)
{
  const int lane = threadIdx.x & 31;
  const int wave = threadIdx.x >> 5;
  const int aRaw = blockIdx.x * 128 + wave * 16 + (lane & 15);
  const int a    = (aRaw < B) ? aRaw : (B - 1);   // keep EXEC full for WMMA

  // per-lane slice of the 3 MLP weight sets (component set widx(t))
  const int half2 = (lane & 16) ? 2 : 0;
  float w1a[8], b1a[8], w2a[8];
  float w1r[8], b1r[8], w2r[8];
  float w1b[8], b1b[8], w2b[8];
#pragma unroll
  for (int t = 0; t < 8; ++t) {
    int wi = 4 * (t >> 1) + (t & 1) + half2;
    w1a[t] = aw1[wi]; b1a[t] = ab1[wi]; w2a[t] = aw2[wi];
    w1r[t] = rw1[wi]; b1r[t] = rb1[wi]; w2r[t] = rw2[wi];
    w1b[t] = bw1[wi]; b1b[t] = bb1[wi]; w2b[t] = bw2[wi];
  }
  const float b2a = ab2[0], b2r = rb2[0], b2b = bb2[0];

  // ego_last cols 1..4 : (px, py, vx, vy) — 16B-aligned float4
  const float4 e4 = *reinterpret_cast<const float4*>(ego + (size_t)a * 68 + 52);
  const float px = e4.x, py = e4.y, vx = e4.z, vy = e4.w;
  const float vn = sqrtf(vx * vx + vy * vy);

  const float p0  = adp[0], p1 = adp[1];
  const float ang = eang[0];
  const float bd0 = border[0], bd3 = border[3];

  // ---- f_destination:  (p1*[|v|,0] - v) / p0, angle-clamped
  float fdx = (p1 * vn - vx) / p0;
  float fdy = (-vy) / p0;
  angle_clamp2(vx, vy, vn, ang, fdx, fdy);

  // ---- neighbors: attraction + repulsion
  const float* nb = nei + (size_t)a * 136;
  float snx = 0.0f, sny = 0.0f;
  for (int j = 0; j < 8; ++j) {
    const float nx  = nb[j * 17 + 1];
    const float ny  = nb[j * 17 + 2];
    const float nvx = nb[j * 17 + 3];
    const float nvy = nb[j * 17 + 4];
    const float rx = nx - px, ry = ny - py;
    const float rn = sqrtf(rx * rx + ry * ry);
    const float dtj = dtf[j];                  // uniform, precomputed

    // attraction: mono_attr(|r|) * r/|r| * dtf[j], per-component mask nei==0
    const float ya  = mono16_wmma(rn, w1a, b1a, w2a, b2a, lane);
    float fax = ya * rx / rn * dtj;
    float fay = ya * ry / rn * dtj;
    if (nx == 0.0f) fax = 0.0f;
    if (ny == 0.0f) fay = 0.0f;
    angle_clamp2(vx, vy, vn, ang, fax, fay);
    snx += fax; sny += fay;

    // repulsion: b = sqrt(|r| + |r+vn*dt|^2 - |vn*dt|^2) / 2
    const float cx = rx + nvx * SFM_DT, cy = ry + nvy * SFM_DT;
    const float wx = nvx * SFM_DT,      wy = nvy * SFM_DT;
    const float bb = rn + (cx * cx + cy * cy) - (wx * wx + wy * wy);
    const float bv = sqrtf(bb) * 0.5f;
    const float yr = mono16_wmma(bv, w1r, b1r, w2r, b2r, lane);
    float frx = yr * rx / rn;
    float fry = yr * ry / rn;
    if (nx == 0.0f) frx = 0.0f;
    if (ny == 0.0f) fry = 0.0f;
    angle_clamp2(vx, vy, vn, ang, frx, fry);
    snx += frx; sny += fry;
  }

  // ---- border: rb = py - border[{0,3}], force vector (0, mono_bor(|rb|)*sign)
  float fby = 0.0f;
#pragma unroll
  for (int i = 0; i < 2; ++i) {
    const float rb = py - (i == 0 ? bd0 : bd3);
    const float rnB = fabsf(rb);
    const float yb = mono16_wmma(rnB, w1b, b1b, w2b, b2b, lane);
    const float fb = yb * (rb / rnB);
    float gx = 0.0f, gy = fb;
    angle_clamp2(vx, vy, vn, ang, gx, gy);
    fby += gy;
  }

  // ---- outputs: [f_dest (B,2) | f_neighbors (B,2) | f_bord (B,2)]
  if (aRaw < B && lane < 16) {
    float* o0 = out;
    float* o1 = out + 2 * (size_t)B;
    float* o2 = out + 4 * (size_t)B;
    o0[2 * aRaw]     = fdx;  o0[2 * aRaw + 1] = fdy;
    o1[2 * aRaw]     = snx;  o1[2 * aRaw + 1] = sny;
    o2[2 * aRaw]     = 0.0f; o2[2 * aRaw + 1] = fby;
  }
}

extern "C" void kernel_launch(void* const* d_in, const int* in_sizes, int n_in,
                              void* d_out, int out_size, void* d_ws, size_t ws_size,
                              hipStream_t stream)
{
  const float* ego  = (const float*)d_in[0];
  const float* nei  = (const float*)d_in[1];
  const float* bord = (const float*)d_in[2];
  const float* adp  = (const float*)d_in[3];
  const float* eang = (const float*)d_in[4];
  const float* aw1 = (const float*)d_in[5],  *ab1 = (const float*)d_in[6];
  const float* aw2 = (const float*)d_in[7],  *ab2 = (const float*)d_in[8];
  const float* rw1 = (const float*)d_in[9],  *rb1 = (const float*)d_in[10];
  const float* rw2 = (const float*)d_in[11], *rb2 = (const float*)d_in[12];
  const float* bw1 = (const float*)d_in[13], *bb1 = (const float*)d_in[14];
  const float* bw2 = (const float*)d_in[15], *bb2 = (const float*)d_in[16];
  const float* dw1 = (const float*)d_in[17], *db1 = (const float*)d_in[18];
  const float* dw2 = (const float*)d_in[19], *db2 = (const float*)d_in[20];

  const int B = in_sizes[0] / 68;   // ego: (B,4,17)
  float* ws = (float*)d_ws;

  sfm_pre<<<1, 32, 0, stream>>>(ego, dw1, db1, dw2, db2, ws);

  dim3 grid((B + 127) / 128);
  sfm_main<<<grid, 256, 0, stream>>>(ego, nei, bord, adp, eang,
                                     aw1, ab1, aw2, ab2,
                                     rw1, rb1, rw2, rb2,
                                     bw1, bb1, bw2, bb2,
                                     ws, (float*)d_out, B);
}